// Radon_30331059044537
// MI455X (gfx1250) — compile-verified
//
#include <hip/hip_runtime.h>
#include <hip/hip_bf16.h>

typedef __attribute__((ext_vector_type(2))) float v2f;
typedef __attribute__((ext_vector_type(8))) float v8f;

#define RW 512
#define RH 512
#define NANG 180

__device__ __forceinline__ float radon_bilin(const float* __restrict__ img,
                                             float ix, float iy) {
    float fx = floorf(ix), fy = floorf(iy);
    float wx1 = ix - fx, wy1 = iy - fy;
    float wx0 = 1.0f - wx1, wy0 = 1.0f - wy1;
    int ix0 = (int)fx, iy0 = (int)fy;
    int ix1 = ix0 + 1, iy1 = iy0 + 1;
    bool vx0 = ((unsigned)ix0 < (unsigned)RW);
    bool vx1 = ((unsigned)ix1 < (unsigned)RW);
    bool vy0 = ((unsigned)iy0 < (unsigned)RH);
    bool vy1 = ((unsigned)iy1 < (unsigned)RH);
    int c0 = min(max(ix0, 0), RW - 1);
    int c1 = min(max(ix1, 0), RW - 1);
    int r0 = min(max(iy0, 0), RH - 1);
    int r1 = min(max(iy1, 0), RH - 1);
    float v00 = img[(r0 << 9) + c0];
    float v01 = img[(r0 << 9) + c1];
    float v10 = img[(r1 << 9) + c0];
    float v11 = img[(r1 << 9) + c1];
    float w00 = (vy0 && vx0) ? wy0 * wx0 : 0.0f;
    float w01 = (vy0 && vx1) ? wy0 * wx1 : 0.0f;
    float w10 = (vy1 && vx0) ? wy1 * wx0 : 0.0f;
    float w11 = (vy1 && vx1) ? wy1 * wx1 : 0.0f;
    return fmaf(v00, w00, fmaf(v01, w01, fmaf(v10, w10, v11 * w11)));
}

// grid = (4, 180, 4) ; block = 256 (8 waves). Each wave -> 16 output columns.
// Each output's 512-term row-sum is split into 4 partials laid out as the
// 16x4 F32 WMMA A-matrix; B = ones(4x16) reduces them on the matrix unit.
__global__ __launch_bounds__(256) void radon_kernel(const float* __restrict__ x,
                                                    const float* __restrict__ theta,
                                                    float* __restrict__ out) {
    const int tid  = threadIdx.x;
    const int lane = tid & 31;
    const int wave = tid >> 5;
    const int m    = lane & 15;   // output slot within wave
    const int half = lane >> 4;   // 0 -> K slots {0,1}; 1 -> K slots {2,3}
    const int ang  = blockIdx.y;
    const int n    = blockIdx.z;
    const int jbase = blockIdx.x * 128 + wave * 16;
    const int j    = jbase + m;

    const float rad = theta[ang] * 0.017453292519943295f;  // fp32, matches ref
    const float c = cosf(rad);
    const float s = sinf(rad);

    // base_j = 2*(j+0.5)/512 - 1  (exact in fp32)
    const float bj = fmaf((float)j + 0.5f, 0.00390625f, -1.0f);
    // ix = bj*c*256 + bi*s*256 + 255.5 ; iy = -bj*s*256 + bi*c*256 + 255.5
    const float ax = fmaf(bj * c, 256.0f, 255.5f);
    const float ay = fmaf(-bj * s, 256.0f, 255.5f);
    const float sx = s * 256.0f;
    const float cy = c * 256.0f;

    const float* img = x + (size_t)n * (RW * RH);

    float acc0 = 0.0f, acc1 = 0.0f;
    const int iA = half * 256;      // K-slot (2*half)   covers i in [iA, iA+128)
    const int iB = iA + 128;        // K-slot (2*half+1) covers i in [iB, iB+128)

    for (int t = 0; t < 128; ++t) {
        {
            float bi = fmaf((float)(iA + t) + 0.5f, 0.00390625f, -1.0f);
            float ix = fmaf(bi, sx, ax);
            float iy = fmaf(bi, cy, ay);
            acc0 += radon_bilin(img, ix, iy);
        }
        {
            float bi = fmaf((float)(iB + t) + 0.5f, 0.00390625f, -1.0f);
            float ix = fmaf(bi, sx, ax);
            float iy = fmaf(bi, cy, ay);
            acc1 += radon_bilin(img, ix, iy);
        }
    }

    // A-matrix (16x4 F32): lane m -> A[m][0..1] = {acc0, acc1};
    //                      lane m+16 -> A[m][2..3] = {acc0, acc1}.
    // B = ones(4x16) (layout-invariant). D[m][*] = sum of the 4 partials.
    v2f a; a.x = acc0; a.y = acc1;
    v2f b; b.x = 1.0f; b.y = 1.0f;
    v8f cacc = {};
    v8f d = __builtin_amdgcn_wmma_f32_16x16x4_f32(
        /*neg_a=*/false, a, /*neg_b=*/false, b,
        /*c_mod=*/(short)0, cacc, /*reuse_a=*/false, /*reuse_b=*/false);

    // D layout: VGPR r: lanes 0-15 hold D[r][lane]; lanes 16-31 hold D[8+r][lane-16].
    // Row m is replicated across N, so lanes 0-7 emit rows 0-7 (d[lane]),
    // lanes 16-23 emit rows 8-15 (d[lane-16]).
    int r = -1, mm = 0;
    if (lane < 8)                    { r = lane;      mm = lane;      }
    else if (lane >= 16 && lane < 24){ r = lane - 16; mm = lane - 8;  }
    if (r >= 0) {
        float val = d[0];
#pragma unroll
        for (int q = 1; q < 8; ++q) val = (r == q) ? d[q] : val;
        // out shape (N, 1, W, NANG)
        out[((size_t)n * RW + (jbase + mm)) * NANG + ang] = val;
    }
}

extern "C" void kernel_launch(void* const* d_in, const int* in_sizes, int n_in,
                              void* d_out, int out_size, void* d_ws, size_t ws_size,
                              hipStream_t stream) {
    const float* x     = (const float*)d_in[0];   // (4,1,512,512) fp32
    const float* theta = (const float*)d_in[1];   // (180,) fp32
    float* out = (float*)d_out;                   // (4,1,512,180) fp32
    (void)in_sizes; (void)n_in; (void)out_size; (void)d_ws; (void)ws_size;
    dim3 grid(4, NANG, 4);   // (j tiles of 128, angle, batch)
    dim3 block(256);
    hipLaunchKernelGGL(radon_kernel, grid, block, 0, stream, x, theta, out);
}